// MultiAgentsSummarizer_14328010900268
// MI455X (gfx1250) — compile-verified
//
#include <hip/hip_runtime.h>

typedef __attribute__((ext_vector_type(2))) float v2f;
typedef __attribute__((ext_vector_type(4))) float v4f;
typedef __attribute__((ext_vector_type(8))) float v8f;

#define B_    8
#define T_    64
#define A_    4
#define S_    512
#define V_    32000
#define EXTV_ 33000

// ---------------------------------------------------------------------------
// Kernel 1: c[bt] = sum_a agent_attn[bt,a] * gen[bt,a]  via V_WMMA_F32_16X16X4_F32
//
// A-matrix (16x4 f32, ISA layout): lanes 0-15 hold {K=0,K=1}, lanes 16-31 hold
// {K=2,K=3}, row M = lane&15.  B-matrix = all ones, so D[i,n] = sum_k A[i,k]
// for EVERY column n -> result independent of B lane layout.  D layout: lanes
// 0-15 VGPR r = D[r, lane], lanes 16-31 VGPR r = D[r+8, lane-16], so lane 0
// holds c[0..7] and lane 16 holds c[8..15] in its 8 accumulator registers.
// ---------------------------------------------------------------------------
__global__ __launch_bounds__(32) void mas_coef_kernel(
    const float* __restrict__ gen, const float* __restrict__ agent_attn,
    float* __restrict__ cbuf) {
  const int lane = threadIdx.x;
  const int bt0  = blockIdx.x * 16;
  const int row  = lane & 15;
  const int kb   = (lane >> 4) << 1;           // 0 for lanes 0-15, 2 for 16-31
  const int base = (bt0 + row) * A_ + kb;

  v2f a;
  a.x = agent_attn[base + 0] * gen[base + 0];
  a.y = agent_attn[base + 1] * gen[base + 1];
  v2f b = {1.0f, 1.0f};
  v8f c = {};
  v8f d = __builtin_amdgcn_wmma_f32_16x16x4_f32(
      /*neg_a=*/false, a, /*neg_b=*/false, b,
      /*c_mod=*/(short)0, c, /*reuse_a=*/false, /*reuse_b=*/false);

  if (lane == 0) {
#pragma unroll
    for (int r = 0; r < 8; ++r) cbuf[bt0 + r] = d[r];
  } else if (lane == 16) {
#pragma unroll
    for (int r = 0; r < 8; ++r) cbuf[bt0 + 8 + r] = d[r];
  }
}

// ---------------------------------------------------------------------------
// Kernel 2: out[bt, 0:V) = c[bt] * vocab[bt, :]; out[bt, V:EXTV) = 0.
// float4 (b128) streaming; vocab loads are non-temporal (read-once) so L2
// capacity stays available for `out`, which the scatter kernel atomically
// updates in L2.
// ---------------------------------------------------------------------------
__global__ __launch_bounds__(256) void mas_scale_kernel(
    const float* __restrict__ vocab, const float* __restrict__ cbuf,
    float* __restrict__ out) {
  const int row = blockIdx.y;                  // bt in [0, B*T)
  const int idx = blockIdx.x * 256 + threadIdx.x;   // float4 index in row
  if (idx >= EXTV_ / 4) return;

  const float c = cbuf[row];
  v4f* orow = reinterpret_cast<v4f*>(out + (size_t)row * EXTV_);
  if (idx < V_ / 4) {
    const v4f* vrow =
        reinterpret_cast<const v4f*>(vocab + (size_t)row * V_);
    v4f v = __builtin_nontemporal_load(&vrow[idx]);
    orow[idx] = c * v;
  } else {
    v4f z = {0.0f, 0.0f, 0.0f, 0.0f};
    orow[idx] = z;
  }
}

// ---------------------------------------------------------------------------
// Kernel 3: scatter-add.  One thread per (b,t,a,s); id = ((b*T+t)*A+a)*S+s so
// the agentwise_attn read is fully coalesced and (bt,a) is uniform per block
// (256 < 512) -> scalar loads for the agent weights.  Atomic f32 adds land in
// the 67.6 MB output, which is resident in the 192 MB L2.
// ---------------------------------------------------------------------------
__global__ __launch_bounds__(256) void mas_scatter_kernel(
    const float* __restrict__ gen, const float* __restrict__ attn,
    const float* __restrict__ agent_attn, const int* __restrict__ article,
    float* __restrict__ out) {
  const int id  = blockIdx.x * 256 + threadIdx.x;  // ((b*T+t)*A+a)*S + s
  const int s   = id & (S_ - 1);
  const int bta = id >> 9;                         // (b*T+t)*A + a
  const int a   = bta & (A_ - 1);
  const int bt  = bta >> 2;
  const int b   = bt / T_;

  const float w   = agent_attn[bta] * (1.0f - gen[bta]);
  const float av  = attn[id];
  const int   tok = article[((b * A_ + a) << 9) + s];

  atomicAdd(out + (size_t)bt * EXTV_ + tok, w * av);
}

extern "C" void kernel_launch(void* const* d_in, const int* in_sizes, int n_in,
                              void* d_out, int out_size, void* d_ws,
                              size_t ws_size, hipStream_t stream) {
  const float* vocab      = (const float*)d_in[0];  // [B,T,V]
  const float* gen        = (const float*)d_in[1];  // [B,T,A]
  const float* attn       = (const float*)d_in[2];  // [B,T,A,S]
  const float* agent_attn = (const float*)d_in[3];  // [B,T,A]
  const int*   article    = (const int*)d_in[4];    // [B,A,S]
  float*       out        = (float*)d_out;          // [B,T,EXT_V]
  float*       cbuf       = (float*)d_ws;           // B*T floats of scratch

  // 1) agent contraction -> c[bt] (32 waves, one WMMA each)
  mas_coef_kernel<<<(B_ * T_) / 16, 32, 0, stream>>>(gen, agent_attn, cbuf);

  // 2) dense scale + tail zero (fully initializes d_out)
  dim3 gridB((EXTV_ / 4 + 255) / 256, B_ * T_);
  mas_scale_kernel<<<gridB, 256, 0, stream>>>(vocab, cbuf, out);

  // 3) copy-probability scatter
  mas_scatter_kernel<<<(B_ * T_ * A_ * S_) / 256, 256, 0, stream>>>(
      gen, attn, agent_attn, article, out);
}